// GraphEncoder_81604378624011
// MI455X (gfx1250) — compile-verified
//
#include <hip/hip_runtime.h>

typedef float v2f __attribute__((ext_vector_type(2)));
typedef float v8f __attribute__((ext_vector_type(8)));

#define N_NODES   50000
#define N_EDGES   600000
#define HIDDEN    128
#define N_GRAPHS  64
#define ND        (N_NODES * HIDDEN)      // 6,400,000
#define EF        (N_EDGES * HIDDEN)      // 76,800,000

// ---------------------------------------------------------------- utilities
__global__ void fill_zero_kernel(float* __restrict__ p, int n) {
    int i = blockIdx.x * blockDim.x + threadIdx.x;
    if (i < n) p[i] = 0.0f;
}

// degree count from dst (one thread per edge)
__global__ void edge_count_kernel(const int* __restrict__ dst, float* __restrict__ degf) {
    int e = blockIdx.x * blockDim.x + threadIdx.x;
    if (e < N_EDGES) atomicAdd(&degf[dst[e]], 1.0f);
}

// dinv = rsqrt(cnt+1); invdeg = 1/(cnt+1); invcnt = 1/max(cnt,1)
__global__ void degree_finalize_kernel(const float* __restrict__ degf,
                                       float* __restrict__ dinv,
                                       float* __restrict__ invdeg,
                                       float* __restrict__ invcnt) {
    int i = blockIdx.x * blockDim.x + threadIdx.x;
    if (i < N_NODES) {
        float c = degf[i];
        float d = c + 1.0f;
        dinv[i]   = rsqrtf(d);
        invdeg[i] = 1.0f / d;
        invcnt[i] = 1.0f / fmaxf(c, 1.0f);
    }
}

// ---------------------------------------------------------------- WMMA GEMM
// C[M,128] = A[M,128] @ W[128,128]  (ACC: C += A @ W)
// One wave per 16x16 tile; block = 256 threads = 8 waves = the 8 N-tiles of
// one M-tile row. grid.x = M/16 = 3125. No divergence -> EXEC all ones.
template <bool ACC>
__global__ void gemm_wmma_kernel(const float* __restrict__ A,
                                 const float* __restrict__ W,
                                 float* __restrict__ C) {
    const int wave   = threadIdx.x >> 5;     // 0..7 -> N tile
    const int lane   = threadIdx.x & 31;
    const int l15    = lane & 15;
    const int hi     = lane >> 4;            // 0: K+{0,1} / M 0..7 ; 1: K+{2,3} / M 8..15
    const int tile_m = blockIdx.x;
    const int col    = wave * 16 + l15;      // output column (N)
    const int arow   = tile_m * 16 + l15;    // A row for this lane's fragment

    v8f acc;
    if (ACC) {
#pragma unroll
        for (int j = 0; j < 8; ++j) {
            int row = tile_m * 16 + j + (hi ? 8 : 0);
            acc[j] = C[row * HIDDEN + col];
        }
    } else {
#pragma unroll
        for (int j = 0; j < 8; ++j) acc[j] = 0.0f;
    }

    const float* arowp = A + arow * HIDDEN;
#pragma unroll 4
    for (int k0 = 0; k0 < HIDDEN; k0 += 4) {
        // A fragment: 16x4, lanes 0-15 K={k0,k0+1}, lanes 16-31 K={k0+2,k0+3}
        v2f a = *(const v2f*)(arowp + k0 + 2 * hi);
        // B fragment: 4x16, same K split, one column per lane
        v2f b;
        b.x = W[(k0 + 2 * hi) * HIDDEN + col];
        b.y = W[(k0 + 2 * hi + 1) * HIDDEN + col];
        acc = __builtin_amdgcn_wmma_f32_16x16x4_f32(false, a, false, b,
                                                    (short)0, acc, false, false);
    }

#pragma unroll
    for (int j = 0; j < 8; ++j) {
        int row = tile_m * 16 + j + (hi ? 8 : 0);
        C[row * HIDDEN + col] = acc[j];
    }
}

// ---------------------------------------------------------------- scatters
// GCN scatter: agg[dst,f] += h[src,f] * dinv[src]*dinv[dst]
__global__ void gcn_scatter_kernel(const float* __restrict__ h,
                                   const int* __restrict__ src,
                                   const int* __restrict__ dst,
                                   const float* __restrict__ dinv,
                                   float* __restrict__ agg) {
    int gid = blockIdx.x * blockDim.x + threadIdx.x;
    if (gid >= EF) return;
    int e = gid >> 7;
    int f = gid & 127;
    int s = src[e];
    int d = dst[e];
    float nrm = dinv[s] * dinv[d];
    atomicAdd(&agg[d * HIDDEN + f], h[s * HIDDEN + f] * nrm);
}

// SAGE scatter (plain sum): agg[dst,f] += h[src,f]
__global__ void sum_scatter_kernel(const float* __restrict__ h,
                                   const int* __restrict__ src,
                                   const int* __restrict__ dst,
                                   float* __restrict__ agg) {
    int gid = blockIdx.x * blockDim.x + threadIdx.x;
    if (gid >= EF) return;
    int e = gid >> 7;
    int f = gid & 127;
    atomicAdd(&agg[dst[e] * HIDDEN + f], h[src[e] * HIDDEN + f]);
}

// out = relu(agg + h * invdeg[i] + b[f])
__global__ void gcn_combine_kernel(const float* __restrict__ agg,
                                   const float* __restrict__ h,
                                   const float* __restrict__ invdeg,
                                   const float* __restrict__ bias,
                                   float* __restrict__ out) {
    int gid = blockIdx.x * blockDim.x + threadIdx.x;
    if (gid >= ND) return;
    int i = gid >> 7;
    int f = gid & 127;
    float v = agg[gid] + h[gid] * invdeg[i] + bias[f];
    out[gid] = fmaxf(v, 0.0f);
}

// buf[i,f] *= invcnt[i]  (neighbor mean)
__global__ void mean_scale_kernel(float* __restrict__ buf,
                                  const float* __restrict__ invcnt) {
    int gid = blockIdx.x * blockDim.x + threadIdx.x;
    if (gid >= ND) return;
    buf[gid] *= invcnt[gid >> 7];
}

// pool accumulate: gacc[batch[i],f] += nodes[i,f] + b_s[f]; gcnt[batch[i]] += 1
__global__ void pool_accum_kernel(const float* __restrict__ nodes,
                                  const int* __restrict__ batch,
                                  const float* __restrict__ bs,
                                  float* __restrict__ gacc,
                                  float* __restrict__ gcnt) {
    int gid = blockIdx.x * blockDim.x + threadIdx.x;
    if (gid >= ND) return;
    int i = gid >> 7;
    int f = gid & 127;
    int g = batch[i];
    atomicAdd(&gacc[g * HIDDEN + f], nodes[gid] + bs[f]);
    if (f == 0) atomicAdd(&gcnt[g], 1.0f);
}

__global__ void pool_final_kernel(const float* __restrict__ gacc,
                                  const float* __restrict__ gcnt,
                                  float* __restrict__ out) {
    int gid = blockIdx.x * blockDim.x + threadIdx.x;
    if (gid >= N_GRAPHS * HIDDEN) return;
    int g = gid >> 7;
    out[gid] = gacc[gid] / fmaxf(gcnt[g], 1.0f);
}

// ---------------------------------------------------------------- launcher
extern "C" void kernel_launch(void* const* d_in, const int* in_sizes, int n_in,
                              void* d_out, int out_size, void* d_ws, size_t ws_size,
                              hipStream_t stream) {
    const float* x    = (const float*)d_in[0];
    const int*   ei   = (const int*)d_in[1];     // [2, E] row-major
    const int*   batch= (const int*)d_in[2];
    const float* W1   = (const float*)d_in[3];
    const float* b1   = (const float*)d_in[4];
    const float* W2   = (const float*)d_in[5];
    const float* b2   = (const float*)d_in[6];
    const float* W_l  = (const float*)d_in[7];
    const float* W_r  = (const float*)d_in[8];
    const float* b_s  = (const float*)d_in[9];
    float* out = (float*)d_out;

    const int* src = ei;            // row 0
    const int* dst = ei + N_EDGES;  // row 1

    float* ws   = (float*)d_ws;
    float* bufA = ws;                         // h_tmp / sage sum
    float* bufB = ws + (size_t)ND;            // agg / features
    float* bufC = ws + (size_t)2 * ND;        // features / sage out
    float* degf   = ws + (size_t)3 * ND;
    float* dinv   = degf + N_NODES;
    float* invdeg = dinv + N_NODES;
    float* invcnt = invdeg + N_NODES;
    float* gacc   = invcnt + N_NODES;         // 64*128
    float* gcnt   = gacc + N_GRAPHS * HIDDEN; // 64

    const int BT = 256;
    const int gN   = (N_NODES + BT - 1) / BT;
    const int gE   = (N_EDGES + BT - 1) / BT;
    const int gND  = (ND + BT - 1) / BT;
    const int gEF  = (EF + BT - 1) / BT;
    const int gPool = (N_GRAPHS * HIDDEN + BT - 1) / BT;

    // ---- degrees (shared by all layers)
    fill_zero_kernel<<<gN, BT, 0, stream>>>(degf, N_NODES);
    fill_zero_kernel<<<(N_GRAPHS * HIDDEN + N_GRAPHS + BT - 1) / BT, BT, 0, stream>>>(
        gacc, N_GRAPHS * HIDDEN + N_GRAPHS);
    edge_count_kernel<<<gE, BT, 0, stream>>>(dst, degf);
    degree_finalize_kernel<<<gN, BT, 0, stream>>>(degf, dinv, invdeg, invcnt);

    // ---- GCN layer 1: bufC = relu(norm-scatter(x@W1) + (x@W1)/deg + b1)
    gemm_wmma_kernel<false><<<N_NODES / 16, 256, 0, stream>>>(x, W1, bufA);
    fill_zero_kernel<<<gND, BT, 0, stream>>>(bufB, ND);
    gcn_scatter_kernel<<<gEF, BT, 0, stream>>>(bufA, src, dst, dinv, bufB);
    gcn_combine_kernel<<<gND, BT, 0, stream>>>(bufB, bufA, invdeg, b1, bufC);

    // ---- GCN layer 2: bufB = relu(...)   (combine writes agg buffer in place)
    gemm_wmma_kernel<false><<<N_NODES / 16, 256, 0, stream>>>(bufC, W2, bufA);
    fill_zero_kernel<<<gND, BT, 0, stream>>>(bufB, ND);
    gcn_scatter_kernel<<<gEF, BT, 0, stream>>>(bufA, src, dst, dinv, bufB);
    gcn_combine_kernel<<<gND, BT, 0, stream>>>(bufB, bufA, invdeg, b2, bufB);

    // ---- SAGE: bufC = mean_neigh(bufB) @ W_l + bufB @ W_r  (+ b_s in pool)
    fill_zero_kernel<<<gND, BT, 0, stream>>>(bufA, ND);
    sum_scatter_kernel<<<gEF, BT, 0, stream>>>(bufB, src, dst, bufA);
    mean_scale_kernel<<<gND, BT, 0, stream>>>(bufA, invcnt);
    gemm_wmma_kernel<false><<<N_NODES / 16, 256, 0, stream>>>(bufA, W_l, bufC);
    gemm_wmma_kernel<true ><<<N_NODES / 16, 256, 0, stream>>>(bufB, W_r, bufC);

    // ---- global mean pool
    pool_accum_kernel<<<gND, BT, 0, stream>>>(bufC, batch, b_s, gacc, gcnt);
    pool_final_kernel<<<gPool, BT, 0, stream>>>(gacc, gcnt, out);
}